// _Memory_12137577578564
// MI455X (gfx1250) — compile-verified
//
#include <hip/hip_runtime.h>
#include <math.h>

#define L_DIM 2048
#define C_DIM 64
#define B_DIM 1024
#define EPSF 1e-16f
#define EPS_COSF 1e-8f
#define LGS 2049   // padded LDS row stride (2048 % 64banks == 0 -> pad by 1)

typedef __attribute__((ext_vector_type(2))) float v2f;
typedef __attribute__((ext_vector_type(8))) float v8f;

// Native fp32 WMMA: D(16x16,f32) = A(16x4,f32) x B(4x16,f32) + C
static __device__ __forceinline__ v8f wmma4(v2f a, v2f b, v8f c) {
  return __builtin_amdgcn_wmma_f32_16x16x4_f32(
      /*neg_a=*/false, a, /*neg_b=*/false, b,
      /*c_mod=*/(short)0, c, /*reuse_a=*/false, /*reuse_b=*/false);
}

static __device__ __forceinline__ float wave_max(float v) {
#pragma unroll
  for (int o = 16; o > 0; o >>= 1) v = fmaxf(v, __shfl_xor(v, o, 32));
  return v;
}
static __device__ __forceinline__ float wave_sum(float v) {
#pragma unroll
  for (int o = 16; o > 0; o >>= 1) v += __shfl_xor(v, o, 32);
  return v;
}

// --- Kernel 1: per-row L2 norms of (seed + EPS) and (key + EPS) ---
__global__ void ntm_norms(const float* __restrict__ seed,
                          const float* __restrict__ key,
                          float* __restrict__ nrm) {
  int t = blockIdx.x * blockDim.x + threadIdx.x;
  if (t < L_DIM) {
    const float* p = seed + t * C_DIM;
    float s = 0.f;
#pragma unroll 8
    for (int c = 0; c < C_DIM; ++c) { float v = p[c] + EPSF; s += v * v; }
    nrm[t] = sqrtf(s);
  } else if (t < L_DIM + B_DIM) {
    const float* p = key + (t - L_DIM) * C_DIM;
    float s = 0.f;
#pragma unroll 8
    for (int c = 0; c < C_DIM; ++c) { float v = p[c] + EPSF; s += v * v; }
    nrm[t] = sqrtf(s);
  }
}

// --- Kernel 2: fused addressing + softmax + shift + pow + output GEMMs ---
// One workgroup handles 16 batch rows; 512 threads = 16 waves.
__global__ __launch_bounds__(512)
void ntm_main(const float* __restrict__ seed, const float* __restrict__ key,
              const float* __restrict__ beta, const float* __restrict__ gate,
              const float* __restrict__ shift, const float* __restrict__ gamma,
              const float* __restrict__ prev, const float* __restrict__ erase,
              const float* __restrict__ addv, const float* __restrict__ nrm,
              float* __restrict__ out) {
  extern __shared__ float sm[];
  float* Lg  = sm;                 // [16][LGS]  logits -> wp -> w
  float* Wg  = Lg + 16 * LGS;      // [16][LGS]  wg; later reduction scratch
  float* Nm  = Wg + 16 * LGS;      // [2048] seed row norms
  float* Kt  = Nm + L_DIM;         // [16][66] key_e tile (padded)
  float* nk  = Kt + 16 * 66;       // [16] key norms
  float* bet = nk + 16;
  float* gat = bet + 16;
  float* gam = gat + 16;
  float* s0a = gam + 16;
  float* s1a = s0a + 16;
  float* s2a = s1a + 16;
  float* qv  = s2a + 16;           // [16] q = sum w^2

  const int tid  = threadIdx.x;
  const int lane = tid & 31;
  const int wv   = tid >> 5;       // wave id 0..15
  const int bt   = blockIdx.x * 16;
  const int mrow = lane & 15;      // lane's M (or N) index inside a 16-wide frag
  const int half = lane >> 4;      // half-wave selects K pair

  // ---- P0: stage shared operands; prefetch the prev_weight HBM stream ----
  {
    const char* pp = (const char*)(prev + (size_t)bt * L_DIM);
    __builtin_prefetch(pp + (size_t)tid * 256, 0, 1);   // 512 x 256B = 128KB tile
  }
  for (int i = tid; i < L_DIM; i += 512) Nm[i] = nrm[i];
  for (int i = tid; i < 16 * C_DIM; i += 512) {
    int m = i >> 6, c = i & 63;
    Kt[m * 66 + c] = key[(bt + m) * C_DIM + c] + EPSF;
  }
  if (tid < 16) {
    int m = tid;
    nk[m]  = nrm[L_DIM + bt + m];
    bet[m] = beta[bt + m];
    gat[m] = gate[bt + m];
    gam[m] = gamma[bt + m];
    s0a[m] = shift[(bt + m) * 3 + 0];
    s1a[m] = shift[(bt + m) * 3 + 1];
    s2a[m] = shift[(bt + m) * 3 + 2];
  }
  __syncthreads();

  // ---- P1: logits = beta * cos(key_e, seed_e) via WMMA (M=16,K=64,N=2048)
  // A = key_e (16x64), B[k][n] = seed_e[l=n][c=k]. Each wave: 8 N-tiles.
  v2f af[16];
#pragma unroll
  for (int ks = 0; ks < 16; ++ks) {
    int c = 4 * ks + 2 * half;
    v2f a; a.x = Kt[mrow * 66 + c]; a.y = Kt[mrow * 66 + c + 1];
    af[ks] = a;
  }
  // per-lane copies of the 8 (m = r + 8*half) scalars used in the epilogue
  float betr[8], nkr[8];
#pragma unroll
  for (int r = 0; r < 8; ++r) {
    int m = r + 8 * half;
    betr[r] = bet[m];
    nkr[r]  = nk[m];
  }
  for (int t = 0; t < 8; ++t) {
    int n0 = (wv * 8 + t) * 16;
    int l  = n0 + mrow;                  // memory slot (column of dot)
    v8f accA = {}, accB = {};            // two chains for XDL ILP
#pragma unroll
    for (int ks = 0; ks < 16; ks += 2) {
      int c0 = 4 * ks + 2 * half;
      const float* sp0 = seed + l * C_DIM + c0;
      v2f b0; b0.x = sp0[0] + EPSF; b0.y = sp0[1] + EPSF;
      accA = wmma4(af[ks], b0, accA);
      int c1 = 4 * (ks + 1) + 2 * half;
      const float* sp1 = seed + l * C_DIM + c1;
      v2f b1; b1.x = sp1[0] + EPSF; b1.y = sp1[1] + EPSF;
      accB = wmma4(af[ks + 1], b1, accB);
    }
    v8f acc = accA + accB;
    float nml = Nm[l];
#pragma unroll
    for (int r = 0; r < 8; ++r) {
      int m = r + 8 * half;
      float rden = __builtin_amdgcn_rcpf(fmaxf(nml * nkr[r], EPS_COSF));
      Lg[m * LGS + l] = betr[r] * acc[r] * rden;
    }
  }
  __syncthreads();

  // ---- P2: softmax over l, blend with prev (one wave per batch row) ----
  {
    const int m = wv;
    float mx = -3.4e38f;
    for (int j = 0; j < 64; ++j) mx = fmaxf(mx, Lg[m * LGS + lane + 32 * j]);
    mx = wave_max(mx);
    float s = 0.f;
    for (int j = 0; j < 64; ++j) s += __expf(Lg[m * LGS + lane + 32 * j] - mx);
    s = wave_sum(s);
    float inv = 1.0f / s;
    float g = gat[m];
    for (int j = 0; j < 64; ++j) {
      int l = lane + 32 * j;
      float wc = __expf(Lg[m * LGS + l] - mx) * inv;
      Wg[m * LGS + l] = g * wc + (1.0f - g) * prev[(bt + m) * L_DIM + l];
    }
  }
  __syncthreads();

  // ---- P3: circular shift conv, ^gamma, normalize; final w -> Lg; q -> qv
  {
    const int m = wv;
    float sh0 = s0a[m], sh1 = s1a[m], sh2 = s2a[m], gm = gam[m];
    float accs = 0.f;
    for (int j = 0; j < 64; ++j) {
      int l  = lane + 32 * j;
      int lm = (l + L_DIM - 1) & (L_DIM - 1);
      int lp = (l + 1) & (L_DIM - 1);
      float wsv = sh0 * Wg[m * LGS + lm] + sh1 * Wg[m * LGS + l] + sh2 * Wg[m * LGS + lp];
      float wp  = __powf(wsv, gm);
      Lg[m * LGS + l] = wp;
      accs += wp;
    }
    accs = wave_sum(accs);
    float inv = 1.0f / (accs + EPSF);
    float q = 0.f;
    for (int j = 0; j < 64; ++j) {
      int l = lane + 32 * j;
      float w = Lg[m * LGS + l] * inv;
      Lg[m * LGS + l] = w;
      q += w * w;
    }
    q = wave_sum(q);
    if (lane == 0) qv[m] = q;
  }
  __syncthreads();

  // ---- P4: S1 = w*seed, S2 = w^2*seed via WMMA (M=16,N=64,K=2048),
  //          K split 4 ways across waves, N split 4 ways. ----
  const int nt = wv & 3;
  const int kc = wv >> 2;
  const int c0 = nt * 16;
  v8f s1v = {}, s2v = {};
  const int kbase = kc * 512;
#pragma unroll 2
  for (int ks = 0; ks < 128; ++ks) {
    int k0 = kbase + 4 * ks;
    int la = k0 + 2 * half;
    float w0 = Lg[mrow * LGS + la];
    float w1 = Lg[mrow * LGS + la + 1];
    v2f a1; a1.x = w0;      a1.y = w1;
    v2f a2; a2.x = w0 * w0; a2.y = w1 * w1;
    const float* sp = seed + la * C_DIM + c0 + mrow;   // B[k][n] = seed[k][c0+n]
    v2f bf; bf.x = sp[0]; bf.y = sp[C_DIM];
    s1v = wmma4(a1, bf, s1v);
    s2v = wmma4(a2, bf, s2v);
  }

  // cross-wave K reduction in LDS (reuse Wg) + epilogue
  float* red = Wg;
  if (kc > 0) {
    int base = ((kc - 1) * 4 + nt) * 512;
#pragma unroll
    for (int r = 0; r < 8; ++r) {
      red[base + r * 32 + lane]       = s1v[r];
      red[base + 256 + r * 32 + lane] = s2v[r];
    }
  }
  __syncthreads();
  if (kc == 0) {
#pragma unroll
    for (int p = 0; p < 3; ++p) {
      int base = (p * 4 + nt) * 512;
#pragma unroll
      for (int r = 0; r < 8; ++r) {
        s1v[r] += red[base + r * 32 + lane];
        s2v[r] += red[base + 256 + r * 32 + lane];
      }
    }
#pragma unroll
    for (int r = 0; r < 8; ++r) {
      int m = r + 8 * half;
      int c = c0 + mrow;
      int gi = (bt + m) * C_DIM + c;
      out[gi] = s1v[r] - erase[gi] * s2v[r] + addv[gi] * qv[m];
    }
  }
}

extern "C" void kernel_launch(void* const* d_in, const int* in_sizes, int n_in,
                              void* d_out, int out_size, void* d_ws, size_t ws_size,
                              hipStream_t stream) {
  const float* seed  = (const float*)d_in[0];
  const float* key   = (const float*)d_in[1];
  const float* beta  = (const float*)d_in[2];
  const float* gate  = (const float*)d_in[3];
  const float* shift = (const float*)d_in[4];
  const float* gamma = (const float*)d_in[5];
  const float* prev  = (const float*)d_in[6];
  const float* erase = (const float*)d_in[7];
  const float* addv  = (const float*)d_in[8];
  float* nrm = (float*)d_ws;           // 3072 floats: [0,2048) seed norms, [2048,3072) key norms
  float* out = (float*)d_out;

  ntm_norms<<<(L_DIM + B_DIM + 255) / 256, 256, 0, stream>>>(seed, key, nrm);

  size_t shmem = (size_t)(2 * 16 * LGS + L_DIM + 16 * 66 + 8 * 16) * sizeof(float);
  ntm_main<<<B_DIM / 16, 512, shmem, stream>>>(seed, key, beta, gate, shift,
                                               gamma, prev, erase, addv, nrm, out);
}